// GraphNeuralNetwork_89550068122357
// MI455X (gfx1250) — compile-verified
//
#include <hip/hip_runtime.h>

typedef __attribute__((ext_vector_type(2))) float v2f;
typedef __attribute__((ext_vector_type(4))) float v4f;
typedef __attribute__((ext_vector_type(8))) float v8f;

#define N_NODES 100000
#define N_EDGES 3200000
#define IN_DIM  128
#define HID     64
#define OUT_DIM 32
#define N_LAYERS 3

// ---------------------------------------------------------------------------
// f32 WMMA tiled GEMM: D[16x16 tile] = (A [+ A2]) x B + bias, optional ReLU.
// All shape params are compile-time so the K-loop fully unrolls into
// immediate-offset global loads + a straight v_wmma_f32_16x16x4_f32 stream.
//
// VGPR layouts per CDNA5 ISA 7.12.2:
//   A 16x4 : row = lane&15 ; lanes 0-15 hold K={0,1}, lanes 16-31 hold K={2,3}
//   B 4x16 : col = lane&15 ; lanes 0-15 hold K={0,1}, lanes 16-31 hold K={2,3}
//   C/D    : col = lane&15 ; row = vgpr + 8*(lane>>4)
// ---------------------------------------------------------------------------
template <int K, int LDN, bool HAS_A2, bool RELU>
__global__ __launch_bounds__(32)
void wmma_gemm_f32_kernel(const float* __restrict__ A,
                          const float* __restrict__ A2,
                          const float* __restrict__ B,
                          const float* __restrict__ bias,
                          float* __restrict__ D)
{
    const int mt   = blockIdx.x;           // 16-row tile
    const int nt   = blockIdx.y;           // 16-col tile
    const int lane = threadIdx.x;
    const int r    = lane & 15;            // A row / B,C,D column within tile
    const int kh   = (lane >> 4) << 1;     // K sub-offset: 0 (lanes 0-15) or 2

    const float* Ap   = A + (size_t)(mt * 16 + r) * K + kh;   // per-lane A base
    const float* A2p  = HAS_A2 ? A2 + (size_t)(mt * 16 + r) * K + kh : nullptr;
    const int    col  = nt * 16 + r;
    const float* Bcol = B + (size_t)kh * LDN + col;           // per-lane B base

    v8f acc = {};
    #pragma unroll
    for (int k0 = 0; k0 < K; k0 += 4) {
        v2f a = *(const v2f*)(Ap + k0);               // one b64 load
        if (HAS_A2) {
            v2f a2 = *(const v2f*)(A2p + k0);
            a.x += a2.x;
            a.y += a2.y;
        }
        v2f b;
        b.x = Bcol[k0 * LDN];                         // immediate offsets
        b.y = Bcol[k0 * LDN + LDN];
        acc = __builtin_amdgcn_wmma_f32_16x16x4_f32(
            /*neg_a=*/false, a, /*neg_b=*/false, b,
            /*c_mod=*/(short)0, acc, /*reuse_a=*/false, /*reuse_b=*/false);
    }

    const float bv    = bias[col];
    float* Dp = D + (size_t)(mt * 16 + ((lane >> 4) << 3)) * LDN + col;
    #pragma unroll
    for (int v = 0; v < 8; ++v) {
        float val = acc[v] + bv;
        if (RELU) val = fmaxf(val, 0.0f);
        Dp[v * LDN] = val;                            // immediate offsets
    }
}

// ---------------------------------------------------------------------------
// Zero the aggregation buffer (float4 stores). Workspace is poisoned 0xAA.
// ---------------------------------------------------------------------------
__global__ __launch_bounds__(256)
void zero_kernel(float4* __restrict__ p, int n4)
{
    int i = blockIdx.x * 256 + threadIdx.x;
    if (i < n4) p[i] = make_float4(0.f, 0.f, 0.f, 0.f);
}

// ---------------------------------------------------------------------------
// Edge scatter: 16 lanes per edge, each lane owns 4 feature dims (16*4 = 64),
// so one wave processes 2 edges with b128 gathers.
//   agg[dst[e]][:] += h[src[e]][:] * w[e]
// h/agg are L2-resident (25.6 MB each on a 192 MB L2); this phase is bound by
// L2 gather bandwidth + f32 atomic throughput, which is the true roofline of
// the whole problem (GEMMs are ~4.5 GFLOP total, negligible).
// ---------------------------------------------------------------------------
__global__ __launch_bounds__(256)
void scatter_kernel(const float* __restrict__ h,
                    const int*   __restrict__ src,
                    const int*   __restrict__ dst,
                    const float* __restrict__ ew,
                    float*       __restrict__ agg)
{
    long long gid  = (long long)blockIdx.x * 256 + threadIdx.x;
    int       e    = (int)(gid >> 4);
    if (e >= N_EDGES) return;
    int       dim4 = (int)(gid & 15) * 4;

    int   s = src[e];          // uniform per half-wave -> single L2 request
    int   d = dst[e];
    float w = ew[e];

    v4f hv = *(const v4f*)(h + (size_t)s * HID + dim4);   // global_load_b128
    float* out = agg + (size_t)d * HID + dim4;
    atomicAdd(out,     hv.x * w);
    atomicAdd(out + 1, hv.y * w);
    atomicAdd(out + 2, hv.z * w);
    atomicAdd(out + 3, hv.w * w);
}

// ---------------------------------------------------------------------------
extern "C" void kernel_launch(void* const* d_in, const int* in_sizes, int n_in,
                              void* d_out, int out_size, void* d_ws, size_t ws_size,
                              hipStream_t stream)
{
    (void)in_sizes; (void)n_in; (void)out_size; (void)ws_size;

    const float* x     = (const float*)d_in[0];
    const int*   ei    = (const int*)  d_in[1];   // [2, E] int32 (x64-disabled JAX)
    const float* ew    = (const float*)d_in[2];
    const float* W_in  = (const float*)d_in[3];
    const float* b_in  = (const float*)d_in[4];
    const float* W_g   = (const float*)d_in[5];   // [L, HID, HID]
    const float* b_g   = (const float*)d_in[6];   // [L, HID]
    const float* W_out = (const float*)d_in[7];
    const float* b_out = (const float*)d_in[8];
    float*       out   = (float*)d_out;

    const int* src = ei;
    const int* dst = ei + N_EDGES;

    float* h   = (float*)d_ws;                     // [N, HID]
    float* agg = h + (size_t)N_NODES * HID;        // [N, HID]

    const dim3 wave(32, 1, 1);
    const dim3 gIn (N_NODES / 16, HID / 16);       // 6250 x 4
    const dim3 gOut(N_NODES / 16, OUT_DIM / 16);   // 6250 x 2

    // h = x @ W_in + b_in
    wmma_gemm_f32_kernel<IN_DIM, HID, false, false>
        <<<gIn, wave, 0, stream>>>(x, nullptr, W_in, b_in, h);

    const int n4 = N_NODES * HID / 4;
    const unsigned scatterBlocks =
        (unsigned)(((long long)N_EDGES * 16 + 255) / 256);

    for (int l = 0; l < N_LAYERS; ++l) {
        zero_kernel<<<(n4 + 255) / 256, 256, 0, stream>>>((float4*)agg, n4);
        scatter_kernel<<<scatterBlocks, 256, 0, stream>>>(h, src, dst, ew, agg);
        // h = relu((h + agg) @ W_g[l] + b_g[l])   (in place: each row tile is shared by HID/16 blocks)
        wmma_gemm_f32_kernel<HID, HID, true, true>
            <<<gIn, wave, 0, stream>>>(h, agg, W_g + (size_t)l * HID * HID,
                                       b_g + (size_t)l * HID, h);
    }

    // out = h @ W_out + b_out
    wmma_gemm_f32_kernel<HID, OUT_DIM, false, false>
        <<<gOut, wave, 0, stream>>>(h, nullptr, W_out, b_out, out);
}